// AdaptiveIzhikevichNeuron_51092930953435
// MI455X (gfx1250) — compile-verified
//
#include <hip/hip_runtime.h>
#include <stdint.h>

// ---------------------------------------------------------------------------
// Izhikevich neuron scan, MI455X (gfx1250).
//   x: [B, C, N, T] fp32, T=32 contiguous. One lane <-> one neuron row (128B).
//   Pure HBM-streaming problem (256 MiB total @ 23.3 TB/s ~ 11.5us floor).
//   Data path: double-buffered GLOBAL_LOAD_ASYNC_TO_LDS_B128 staging
//   (ASYNCcnt / s_wait_asynccnt), padded LDS rows read via native DS path
//   (addrspace(3) -> ds_load_b128), non-temporal B128 stores.
// ---------------------------------------------------------------------------

typedef float f32x4 __attribute__((ext_vector_type(4)));
// LDS-resident vector pointer: forces the native DS path (ds_load_b128),
// not FLAT-via-src_shared_base.
typedef __attribute__((address_space(3))) const f32x4* lds_row_t;

#define WAVES_PER_BLOCK 4
#define BLOCK_THREADS   (WAVES_PER_BLOCK * 32)
#define TILE_NEURONS    32                       // one wave's tile
#define T_STEPS         32
#define TILE_BYTES      (TILE_NEURONS * T_STEPS * 4)  // 4096 B
#define ROW_DW          36                       // 32 data dwords + 4 pad (144 B row)
#define ROW_BYTES       (ROW_DW * 4)             // 144 B
#define BUF_DW          (TILE_NEURONS * ROW_DW)  // 1152 dwords = 4608 B per buffer
#define BUF_BYTES       (BUF_DW * 4)
#define NUM_BLOCKS      2048

// Stage one 32-neuron tile (4 KB) global -> LDS via async copy.
// Global: coalesced, chunk s = i*32+lane covers bytes [s*16, s*16+16).
// LDS:    row-padded layout, chunk s lands at (s>>3)*144 + (s&7)*16.
// For fixed lane: global advances 512 B per i, LDS advances 576 B per i.
__device__ __forceinline__ void stage_tile_async(const char* xbase, int tile,
                                                 unsigned ldsBufByte, int lane) {
    uint64_t g = (uint64_t)xbase + (uint64_t)tile * TILE_BYTES + (unsigned)lane * 16u;
    unsigned  l = ldsBufByte + (unsigned)(lane >> 3) * 144u + (unsigned)(lane & 7) * 16u;
#pragma unroll
    for (int i = 0; i < 8; ++i) {
        asm volatile("global_load_async_to_lds_b128 %0, %1, off"
                     :
                     : "v"(l + (unsigned)i * 576u), "v"(g + (uint64_t)i * 512u)
                     : "memory");
    }
}

__global__ __launch_bounds__(BLOCK_THREADS, 1)
void izhikevich_scan_kernel(const float* __restrict__ x,
                            const float* __restrict__ pa,
                            const float* __restrict__ pb,
                            const float* __restrict__ pc,
                            const float* __restrict__ pd,
                            float* __restrict__ out,
                            int nTiles, int waveStride) {
    __shared__ __align__(16) float lds[WAVES_PER_BLOCK * 2 * BUF_DW];

    const int lane = threadIdx.x & 31;
    const int wid  = threadIdx.x >> 5;
    const int gw   = blockIdx.x * WAVES_PER_BLOCK + wid;   // global wave id

    // Uniform scalar parameters (s_load through constant path).
    const float A  = pa[0];
    const float Bp = pb[0];
    const float Cc = pc[0];
    const float Dd = pd[0];

    // Low 32 bits of the flat shared address == LDS byte offset on CDNA5
    // (aperture lives in the high 32 bits). All LDS addressing below is done
    // with these 32-bit offsets, both for the async copies and the DS reads.
    const unsigned ldsBase = (unsigned)(uintptr_t)&lds[0];
    unsigned bufOff[2];
    bufOff[0] = ldsBase + (unsigned)(wid * 2) * BUF_BYTES;
    bufOff[1] = bufOff[0] + BUF_BYTES;

    int tile = gw;
    if (tile >= nTiles) return;

    // Prime the pipeline: tile 0 of this wave into buffer 0.
    stage_tile_async((const char*)x, tile, bufOff[0], lane);

    int cur = 0;
    for (; tile < nTiles; tile += waveStride, cur ^= 1) {
        const int  nextTile = tile + waveStride;
        const bool hasNext  = nextTile < nTiles;

        if (hasNext) {
            // WAR hygiene: the buffer we are about to overwrite was fully read
            // in the previous iteration; make sure those DS reads retired
            // before async writes can land in it.
            asm volatile("s_wait_dscnt 0x0" ::: "memory");
            stage_tile_async((const char*)x, nextTile, bufOff[cur ^ 1], lane);
            // 8 async ops for current tile + 8 for next are in flight; async
            // loads complete in order, so <=8 outstanding => current resident.
            asm volatile("s_wait_asynccnt 0x8" ::: "memory");
        } else {
            asm volatile("s_wait_asynccnt 0x0" ::: "memory");
        }

        // This lane's neuron row in LDS (144 B stride => 2-way conflict max).
        lds_row_t row = (lds_row_t)(bufOff[cur] + (unsigned)lane * ROW_BYTES);
        f32x4* dst = (f32x4*)((char*)out + (uint64_t)tile * TILE_BYTES
                                         + (unsigned)lane * 128u);

        float v = 0.0f;
        float u = 0.0f;
#pragma unroll
        for (int j = 0; j < 8; ++j) {
            f32x4 xv = row[j];                 // ds_load_b128, imm offset j*16
#pragma unroll
            for (int e = 0; e < 4; ++e) {
                const float xt = xv[e];
                // dv = 0.04 v^2 + 5 v + 140 - u + x
                float dv = fmaf(0.04f * v, v, fmaf(5.0f, v, 140.0f)) - u + xt;
                v = v + dv;
                // u += a * (b * v_new - u)
                u = fmaf(A, fmaf(Bp, v, -u), u);
                const bool  fire  = (v >= 30.0f);
                const float spike = fire ? 1.0f : 0.0f;
                v = fire ? Cc : v;             // v*(1-s) + c*s
                u = fmaf(Dd, spike, u);        // u + d*s
                xv[e] = spike;
            }
            // Streaming output: bypass cache retention (th:NT).
            __builtin_nontemporal_store(xv, dst + j);
        }
    }
}

extern "C" void kernel_launch(void* const* d_in, const int* in_sizes, int n_in,
                              void* d_out, int out_size, void* d_ws, size_t ws_size,
                              hipStream_t stream) {
    const float* x  = (const float*)d_in[0];
    const float* pa = (const float*)d_in[1];
    const float* pb = (const float*)d_in[2];
    const float* pc = (const float*)d_in[3];
    const float* pd = (const float*)d_in[4];
    float*       o  = (float*)d_out;

    const int nNeurons   = in_sizes[0] / T_STEPS;        // B*C*N
    const int nTiles     = nNeurons / TILE_NEURONS;      // 32 neurons per wave-tile
    const int waveStride = NUM_BLOCKS * WAVES_PER_BLOCK;

    izhikevich_scan_kernel<<<dim3(NUM_BLOCKS), dim3(BLOCK_THREADS), 0, stream>>>(
        x, pa, pb, pc, pd, o, nTiles, waveStride);
}